// GCN_3917010174011
// MI455X (gfx1250) — compile-verified
//
#include <hip/hip_runtime.h>

#define N_NODES 100000
#define N_FEAT  512
#define HIDDEN  16
#define OUT_F   2
#define N_EDGES 3200000

typedef __attribute__((ext_vector_type(2))) float v2f;
typedef __attribute__((ext_vector_type(8))) float v8f;

// ---------------- degree / normalization ----------------
__global__ void deg_init(float* __restrict__ deg) {
    int i = blockIdx.x * blockDim.x + threadIdx.x;
    if (i < N_NODES) deg[i] = 1.0f;   // self loop
}

__global__ void deg_scatter(const long long* __restrict__ ei, float* __restrict__ deg) {
    int e = blockIdx.x * blockDim.x + threadIdx.x;
    if (e < N_EDGES) {
        int dst = (int)ei[N_EDGES + e];
        unsafeAtomicAdd(&deg[dst], 1.0f);
    }
}

__global__ void deg_rsqrt(float* __restrict__ deg) {
    int i = blockIdx.x * blockDim.x + threadIdx.x;
    if (i < N_NODES) deg[i] = rsqrtf(deg[i]);   // becomes dinv
}

// ---------------- layer-1 GEMM: H = X @ W1  (f32 WMMA) ----------------
// One wave computes a 16x16 output tile (16 rows, HIDDEN=16 cols),
// accumulating K=512 in steps of 4 via v_wmma_f32_16x16x4_f32.
__global__ void __launch_bounds__(256) gemm1_wmma(const float* __restrict__ X,
                                                  const float* __restrict__ W1,
                                                  float* __restrict__ H) {
    __shared__ float w1s[N_FEAT * HIDDEN];   // 32 KB
    for (int i = threadIdx.x; i < N_FEAT * HIDDEN; i += 256) w1s[i] = W1[i];
    __syncthreads();

    const int wave = threadIdx.x >> 5;
    const int lane = threadIdx.x & 31;
    const int tile = blockIdx.x * 8 + wave;
    if (tile * 16 >= N_NODES) return;        // uniform per wave -> EXEC all-1s inside

    const int half = lane >> 4;              // 0: K={0,1}, 1: K={2,3} per A/B layout
    const int r    = lane & 15;              // row (A) / col (B,D)

    const float* xrow = X + (size_t)(tile * 16 + r) * N_FEAT;
    v8f acc = {};

#pragma unroll 8
    for (int k = 0; k < N_FEAT; k += 4) {
        const int kk = k + 2 * half;
        v2f a = *(const v2f*)(xrow + kk);                       // global_load_b64
        v2f b;
        b.x = w1s[kk * HIDDEN + r];                             // ds_load_b32
        b.y = w1s[(kk + 1) * HIDDEN + r];
        acc = __builtin_amdgcn_wmma_f32_16x16x4_f32(
            /*neg_a=*/false, a, /*neg_b=*/false, b,
            /*c_mod=*/(short)0, acc, /*reuse_a=*/false, /*reuse_b=*/false);
    }

    float* hout = H + (size_t)tile * 16 * HIDDEN;
#pragma unroll
    for (int v = 0; v < 8; ++v)
        hout[(v + 8 * half) * HIDDEN + r] = acc[v];             // D layout: VGPR v -> M=v(+8)
}

// ---------------- layer-1 aggregation ----------------
__global__ void self_init1(const float* __restrict__ H, const float* __restrict__ dinv,
                           const float* __restrict__ b1, float* __restrict__ acc1) {
    int i = blockIdx.x * blockDim.x + threadIdx.x;
    if (i < N_NODES * HIDDEN) {
        int n = i >> 4, f = i & 15;
        float d = dinv[n];
        acc1[i] = H[i] * d * d + b1[f];
    }
}

__global__ void scatter1(const long long* __restrict__ ei, const float* __restrict__ H,
                         const float* __restrict__ dinv, float* __restrict__ acc1) {
    int gid = blockIdx.x * blockDim.x + threadIdx.x;            // < 51.2M, fits int
    if (gid >= N_EDGES * HIDDEN) return;
    int f = gid & (HIDDEN - 1);
    int e = gid >> 4;
    int src = (int)ei[e];
    int dst = (int)ei[N_EDGES + e];
    float norm = dinv[src] * dinv[dst];
    unsafeAtomicAdd(&acc1[dst * HIDDEN + f], H[src * HIDDEN + f] * norm);
}

// ---------------- relu + layer-2 GEMM (tiny) + self term ----------------
__global__ void mid_relu_gemm2(const float* __restrict__ acc1, const float* __restrict__ W2,
                               const float* __restrict__ b2, const float* __restrict__ dinv,
                               float* __restrict__ g, float* __restrict__ out) {
    int n = blockIdx.x * blockDim.x + threadIdx.x;
    if (n >= N_NODES) return;
    const float* a = acc1 + n * HIDDEN;
    float g0 = 0.f, g1 = 0.f;
#pragma unroll
    for (int f = 0; f < HIDDEN; ++f) {
        float h = a[f];
        h = h > 0.f ? h : 0.f;                                  // relu (b1 already folded in)
        g0 += h * W2[f * OUT_F + 0];
        g1 += h * W2[f * OUT_F + 1];
    }
    float d = dinv[n], s = d * d;
    g[n * OUT_F + 0] = g0;
    g[n * OUT_F + 1] = g1;
    out[n * OUT_F + 0] = g0 * s + b2[0];                        // self-loop term + bias
    out[n * OUT_F + 1] = g1 * s + b2[1];
}

__global__ void scatter2(const long long* __restrict__ ei, const float* __restrict__ g,
                         const float* __restrict__ dinv, float* __restrict__ out) {
    int e = blockIdx.x * blockDim.x + threadIdx.x;
    if (e >= N_EDGES) return;
    int src = (int)ei[e];
    int dst = (int)ei[N_EDGES + e];
    float norm = dinv[src] * dinv[dst];
    v2f gv = *(const v2f*)(g + src * OUT_F);
    unsafeAtomicAdd(&out[dst * OUT_F + 0], gv.x * norm);
    unsafeAtomicAdd(&out[dst * OUT_F + 1], gv.y * norm);
}

// ---------------- launcher ----------------
extern "C" void kernel_launch(void* const* d_in, const int* in_sizes, int n_in,
                              void* d_out, int out_size, void* d_ws, size_t ws_size,
                              hipStream_t stream) {
    const float*     x  = (const float*)d_in[0];
    const long long* ei = (const long long*)d_in[1];   // int64 [2, E]
    const float*     W1 = (const float*)d_in[2];
    const float*     b1 = (const float*)d_in[3];
    const float*     W2 = (const float*)d_in[4];
    const float*     b2 = (const float*)d_in[5];
    float* out = (float*)d_out;

    float* ws   = (float*)d_ws;
    float* dinv = ws;                                   // 100000
    float* H    = ws + N_NODES;                         // 1.6M
    float* acc1 = H + (size_t)N_NODES * HIDDEN;         // 1.6M
    float* g    = acc1 + (size_t)N_NODES * HIDDEN;      // 200000

    const int T = 256;
    deg_init   <<<(N_NODES + T - 1) / T, T, 0, stream>>>(dinv);
    deg_scatter<<<(N_EDGES + T - 1) / T, T, 0, stream>>>(ei, dinv);
    deg_rsqrt  <<<(N_NODES + T - 1) / T, T, 0, stream>>>(dinv);

    const int n_tiles = (N_NODES + 15) / 16;            // 6250
    gemm1_wmma <<<(n_tiles + 7) / 8, T, 0, stream>>>(x, W1, H);

    self_init1 <<<(N_NODES * HIDDEN + T - 1) / T, T, 0, stream>>>(H, dinv, b1, acc1);
    scatter1   <<<(N_EDGES * HIDDEN + T - 1) / T, T, 0, stream>>>(ei, H, dinv, acc1);

    mid_relu_gemm2<<<(N_NODES + T - 1) / T, T, 0, stream>>>(acc1, W2, b2, dinv, g, out);
    scatter2   <<<(N_EDGES + T - 1) / T, T, 0, stream>>>(ei, g, dinv, out);
}